// AuxNet_37460704756355
// MI455X (gfx1250) — compile-verified
//
#include <hip/hip_runtime.h>
#include <math.h>

// ---------------- problem constants ----------------
#define A_N    2048          // aux experts
#define HN     256           // hidden width
#define OUTN   2
#define BLAY   4
#define ELAY   4
#define NLAY   (BLAY + A_N + ELAY + 1)   // 2057
#define KTOT   ((A_N + 1) * HN)          // 524544 rows of We_first
#define MVB    683                        // split-K blocks
#define MVS    192                        // K/4 steps per block (683*192*4 == 524544)

// ---------------- workspace layout (floats) ----------------
#define WS_XEND   0u                               // 524544: scaled end-input (un-normalized by gsum)
#define WS_PART   524544u                          // 683*256 partial sums
#define WS_AUX    (WS_PART + MVB * HN)             // 2048*2 aux softmax outputs
#define WS_SCAL   (WS_AUX + A_N * OUTN)            // [0]=gsum, [1]=wsum
#define WS_HEADS  (WS_SCAL + 2u)                   // base outs (8) + middle_out (2)

typedef __attribute__((ext_vector_type(2))) float v2f;
typedef __attribute__((ext_vector_type(8))) float v8f;

// ---------------- helpers ----------------
__device__ __forceinline__ void tree2(float* r0, float* r1, int tid) {
  for (int s = 128; s > 0; s >>= 1) {
    __syncthreads();
    if (tid < s) { r0[tid] += r0[tid + s]; r1[tid] += r1[tid + s]; }
  }
  __syncthreads();
}

// softmax head over OUT=2: out2 = softmax(xc . Wo + bo); xc is shared[256]
__device__ __forceinline__ void head2(const float* xc, const float* __restrict__ Wo,
                                      const float* __restrict__ bo,
                                      float* r0, float* r1, int tid, float* out2) {
  r0[tid] = xc[tid] * Wo[tid * OUTN + 0];
  r1[tid] = xc[tid] * Wo[tid * OUTN + 1];
  tree2(r0, r1, tid);
  if (tid == 0) {
    float z0 = r0[0] + bo[0], z1 = r1[0] + bo[1];
    float mx = fmaxf(z0, z1);
    float e0 = expf(z0 - mx), e1 = expf(z1 - mx);
    float inv = 1.0f / (e0 + e1);
    out2[0] = e0 * inv; out2[1] = e1 * inv;
  }
  __syncthreads();
}

// ---------------- kernel 1: base stack + middle + scalar sums ----------------
__global__ __launch_bounds__(256) void k1_base(
    const float* __restrict__ base_feature,
    const float* __restrict__ Wb_first, const float* __restrict__ bb_first,
    const float* __restrict__ Wb_rest,  const float* __restrict__ bb_rest,
    const float* __restrict__ Wbo,      const float* __restrict__ bbo,
    const float* __restrict__ Wm,       const float* __restrict__ bm,
    const float* __restrict__ Wmo,      const float* __restrict__ bmo,
    const int*   __restrict__ Xmask,
    const float* __restrict__ lw,
    float* __restrict__ ws) {
  __shared__ float xc[HN], r0[HN], r1[HN];
  const int j = threadIdx.x;

  // first layer: IN=1
  float bf = base_feature[0];
  xc[j] = fmaxf(bf * Wb_first[j] + bb_first[j], 0.0f);
  __syncthreads();
  head2(xc, Wbo + 0 * HN * OUTN, bbo + 0 * OUTN, r0, r1, j, ws + WS_HEADS + 0);

  // 3 more base layers
  for (int i = 0; i < BLAY - 1; ++i) {
    float s = bb_rest[i * HN + j];
    const float* W = Wb_rest + (size_t)i * HN * HN;
    for (int k = 0; k < HN; ++k) s += xc[k] * W[k * HN + j];
    __syncthreads();
    xc[j] = fmaxf(s, 0.0f);
    __syncthreads();
    head2(xc, Wbo + (i + 1) * HN * OUTN, bbo + (i + 1) * OUTN, r0, r1, j,
          ws + WS_HEADS + 2 * (i + 1));
  }

  // middle layer
  {
    float s = bm[j];
    for (int k = 0; k < HN; ++k) s += xc[k] * Wm[k * HN + j];
    __syncthreads();
    float mid = fmaxf(s, 0.0f);
    xc[j] = mid;
    // middle row of end-input, pre-scaled by its gamma weight (normalization deferred)
    ws[WS_XEND + (size_t)A_N * HN + j] = mid * lw[NLAY - 1];
    __syncthreads();
    head2(xc, Wmo, bmo, r0, r1, j, ws + WS_HEADS + 8);
  }

  // gsum / wsum (deterministic tree reduction of masked aux weights)
  float pg = 0.0f;
  for (int a = j; a < A_N; a += HN)
    pg += (Xmask[a] != 0) ? lw[BLAY + a] : 0.0f;
  r0[j] = pg; r1[j] = 0.0f;
  tree2(r0, r1, j);
  if (j == 0) {
    float maskedAux = r0[0];
    float gsum = maskedAux + lw[NLAY - 1];
    float always = 0.0f;
    for (int i = 0; i < BLAY; ++i) always += lw[i];
    for (int i = BLAY + A_N; i < NLAY; ++i) always += lw[i];
    ws[WS_SCAL + 0] = gsum;
    ws[WS_SCAL + 1] = maskedAux + always;  // wsum
  }
}

// ---------------- kernel 2: aux experts ----------------
__global__ __launch_bounds__(256) void k2_aux(
    const float* __restrict__ X, const int* __restrict__ Xmask,
    const float* __restrict__ Wa, const float* __restrict__ ba,
    const float* __restrict__ Wao, const float* __restrict__ bao,
    const float* __restrict__ lw,
    float* __restrict__ ws) {
  __shared__ float r0[HN], r1[HN];
  const int a = blockIdx.x, j = threadIdx.x;
  const float xa = X[a];
  const float m  = (Xmask[a] != 0) ? 1.0f : 0.0f;
  const float h  = fmaxf(xa * Wa[(size_t)a * HN + j] + ba[(size_t)a * HN + j], 0.0f);

  ws[WS_XEND + (size_t)a * HN + j] = h * m * lw[BLAY + a];

  r0[j] = h * Wao[((size_t)a * HN + j) * OUTN + 0];
  r1[j] = h * Wao[((size_t)a * HN + j) * OUTN + 1];
  tree2(r0, r1, j);
  if (j == 0) {
    float z0 = r0[0] + bao[a * OUTN + 0], z1 = r1[0] + bao[a * OUTN + 1];
    float mx = fmaxf(z0, z1);
    float e0 = expf(z0 - mx), e1 = expf(z1 - mx);
    float inv = 1.0f / (e0 + e1);
    ws[WS_AUX + a * OUTN + 0] = e0 * inv;
    ws[WS_AUX + a * OUTN + 1] = e1 * inv;
  }
}

// ---------------- kernel 3: 537MB matvec via V_WMMA_F32_16X16X4_F32 ----------------
// Wave w owns output columns [16w, 16w+16); block b owns K4 steps [b*MVS, (b+1)*MVS).
// A-matrix (16x4 f32): VGPR0 = K0 (lanes 0-15) / K2 (lanes 16-31), VGPR1 = K1/K3.
// x replicated across all 16 M rows -> every D row equals the matvec partial.
// B-matrix (4x16 f32) mirrors that K mapping; N = lane&15.
// D row M=0 lives on VGPR0 lanes 0-15 -> c[0].
//
// Base pointers fixed, indexed by s*const so the backend strength-reduces to
// immediate-offset loads (step stride 4KB << 24-bit IOFFSET range).
// Two accumulators break the WMMA C-dependency chain (deterministic: even/odd
// step sums added once at the store). We_first is a single-use 537MB stream ->
// non-temporal loads keep xend + partials hot in the 192MB L2.
__global__ __launch_bounds__(512) void k3_matvec(
    const float* __restrict__ W,       // We_first [KTOT, HN]
    const float* __restrict__ xend,    // scaled end-input [KTOT]
    float* __restrict__ partial) {     // [MVB, HN]
  const int tid  = threadIdx.x;
  const int wave = tid >> 5;
  const int lane = tid & 31;
  const int jt   = wave << 4;
  const int j    = jt + (lane & 15);
  const int off0 = (lane >> 4) << 1;   // K offset for VGPR0 lane-half: 0 or 2
  const int kbase = blockIdx.x * (MVS * 4);

  const float* __restrict__ pA  = xend + kbase + off0;               // {pA[4s], pA[4s+1]}
  const float* __restrict__ pB0 = W + (size_t)(kbase + off0) * HN + j;
  const float* __restrict__ pB1 = pB0 + HN;
  const bool pf_ok = (blockIdx.x != MVB - 1);

  v8f c0 = {}, c1 = {};
  #pragma unroll 4
  for (int s = 0; s < MVS; ++s) {
    if (pf_ok && (s & 15) == 0)
      __builtin_prefetch(pB0 + (size_t)(s + 16) * 4 * HN, 0, 1);  // global_prefetch_b8, 64 rows ahead
    v2f a, b;
    a.x = pA[s * 4 + 0];
    a.y = pA[s * 4 + 1];
    b.x = __builtin_nontemporal_load(pB0 + (size_t)s * 4 * HN);
    b.y = __builtin_nontemporal_load(pB1 + (size_t)s * 4 * HN);
    if (s & 1)
      c1 = __builtin_amdgcn_wmma_f32_16x16x4_f32(
          /*neg_a=*/false, a, /*neg_b=*/false, b,
          /*c_mod=*/(short)0, c1, /*reuse_a=*/false, /*reuse_b=*/false);
    else
      c0 = __builtin_amdgcn_wmma_f32_16x16x4_f32(
          /*neg_a=*/false, a, /*neg_b=*/false, b,
          /*c_mod=*/(short)0, c0, /*reuse_a=*/false, /*reuse_b=*/false);
  }
  if (lane < 16)
    partial[(size_t)blockIdx.x * HN + jt + lane] = c0[0] + c1[0];
}

// ---------------- kernel 4: reduce + end stack + hedge combine ----------------
__global__ __launch_bounds__(256) void k4_end(
    const float* __restrict__ be_first,
    const float* __restrict__ We_rest, const float* __restrict__ be_rest,
    const float* __restrict__ Weo,     const float* __restrict__ beo,
    const int*   __restrict__ Xmask,
    const float* __restrict__ lw,
    const float* __restrict__ ws,
    float* __restrict__ out) {
  __shared__ float xc[HN], r0[HN], r1[HN], eo[ELAY * OUTN];
  const int j = threadIdx.x;

  // deterministic reduction of split-K partials
  float y = 0.0f;
  for (int b = 0; b < MVB; ++b) y += ws[WS_PART + (size_t)b * HN + j];
  const float gsum = ws[WS_SCAL + 0];
  const float wsum = ws[WS_SCAL + 1];
  xc[j] = fmaxf(y / gsum + be_first[j], 0.0f);
  __syncthreads();
  head2(xc, Weo + 0, beo + 0, r0, r1, j, eo + 0);

  for (int i = 0; i < ELAY - 1; ++i) {
    float s = be_rest[i * HN + j];
    const float* W = We_rest + (size_t)i * HN * HN;
    for (int k = 0; k < HN; ++k) s += xc[k] * W[k * HN + j];
    __syncthreads();
    xc[j] = fmaxf(s, 0.0f);
    __syncthreads();
    head2(xc, Weo + (i + 1) * HN * OUTN, beo + (i + 1) * OUTN, r0, r1, j,
          eo + 2 * (i + 1));
  }

  // hedge-weighted combine: layer order = base(4), aux(2048), end(4), middle(1)
  const float invw = 1.0f / wsum;
  float a0 = 0.0f, a1 = 0.0f;
  for (int i = j; i < NLAY; i += HN) {
    float l0, l1, act;
    if (i < BLAY)                 { l0 = ws[WS_HEADS + 2 * i];            l1 = ws[WS_HEADS + 2 * i + 1];            act = 1.0f; }
    else if (i < BLAY + A_N)      { int a = i - BLAY;
                                    l0 = ws[WS_AUX + a * OUTN];           l1 = ws[WS_AUX + a * OUTN + 1];
                                    act = (Xmask[a] != 0) ? 1.0f : 0.0f; }
    else if (i < BLAY + A_N+ELAY) { int e = i - BLAY - A_N;
                                    l0 = eo[e * OUTN];                    l1 = eo[e * OUTN + 1];                    act = 1.0f; }
    else                          { l0 = ws[WS_HEADS + 8];                l1 = ws[WS_HEADS + 9];                    act = 1.0f; }
    const float wi = lw[i] * act * invw;
    const float o0 = l0 * wi, o1 = l1 * wi;
    out[2 + i * OUTN + 0] = o0;
    out[2 + i * OUTN + 1] = o1;
    a0 += o0; a1 += o1;
  }
  r0[j] = a0; r1[j] = a1;
  tree2(r0, r1, j);
  if (j == 0) { out[0] = r0[0]; out[1] = r1[0]; }
}

// ---------------- launch ----------------
extern "C" void kernel_launch(void* const* d_in, const int* in_sizes, int n_in,
                              void* d_out, int out_size, void* d_ws, size_t ws_size,
                              hipStream_t stream) {
  const float* X            = (const float*)d_in[0];
  const int*   X_mask       = (const int*)  d_in[1];
  const float* base_feature = (const float*)d_in[2];
  const float* Wb_first     = (const float*)d_in[3];
  const float* bb_first     = (const float*)d_in[4];
  const float* Wb_rest      = (const float*)d_in[5];
  const float* bb_rest      = (const float*)d_in[6];
  const float* Wbo          = (const float*)d_in[7];
  const float* bbo          = (const float*)d_in[8];
  const float* Wm           = (const float*)d_in[9];
  const float* bm           = (const float*)d_in[10];
  const float* Wmo          = (const float*)d_in[11];
  const float* bmo          = (const float*)d_in[12];
  const float* Wa           = (const float*)d_in[13];
  const float* ba           = (const float*)d_in[14];
  const float* Wao          = (const float*)d_in[15];
  const float* bao          = (const float*)d_in[16];
  const float* We_first     = (const float*)d_in[17];
  const float* be_first     = (const float*)d_in[18];
  const float* We_rest      = (const float*)d_in[19];
  const float* be_rest      = (const float*)d_in[20];
  const float* Weo          = (const float*)d_in[21];
  const float* beo          = (const float*)d_in[22];
  const float* lw           = (const float*)d_in[23];

  float* ws  = (float*)d_ws;
  float* out = (float*)d_out;

  k1_base<<<1, 256, 0, stream>>>(base_feature, Wb_first, bb_first, Wb_rest, bb_rest,
                                 Wbo, bbo, Wm, bm, Wmo, bmo, X_mask, lw, ws);
  k2_aux<<<A_N, 256, 0, stream>>>(X, X_mask, Wa, ba, Wao, bao, lw, ws);
  k3_matvec<<<MVB, 512, 0, stream>>>(We_first, ws + WS_XEND, ws + WS_PART);
  k4_end<<<1, 256, 0, stream>>>(be_first, We_rest, be_rest, Weo, beo,
                                X_mask, lw, ws, out);
}